// SparseGatedMLP_32676111188159
// MI455X (gfx1250) — compile-verified
//
#include <hip/hip_runtime.h>
#include <hip/hip_bf16.h>

typedef __attribute__((ext_vector_type(2))) float v2f;
typedef __attribute__((ext_vector_type(8))) float v8f;

#define D_IN    512
#define D_OUT   512
#define R_ROWS  32
#define BLOCK   256
#define VSTRIDE 520   // 512 + 8 floats pad: 2*VSTRIDE % 64 != 0 -> no LDS bank conflicts

// dynamic LDS layout (floats):
//   [0, 32*VSTRIDE)          vrows  : staged out_weight rows (async-copied)
//   [VR_F, VR_F+512)         xs     : token vector
//   [.. +32)                 coefs
//   [.. +32)                 gates
//   [.. +32)                 idxs (int)
#define VR_F   (R_ROWS * VSTRIDE)
#define SMEM_F (VR_F + D_IN + 3 * R_ROWS)

__global__ __launch_bounds__(BLOCK) void sparse_gated_mlp_kernel(
    const float* __restrict__ x,
    const int*   __restrict__ indices,
    const float* __restrict__ in_w,
    const float* __restrict__ gate_w,
    const float* __restrict__ out_w,
    float*       __restrict__ out,
    int n_tok)
{
    extern __shared__ float smem[];
    float* vrows = smem;
    float* xs    = smem + VR_F;
    float* coefs = xs + D_IN;
    float* gates = coefs + R_ROWS;
    int*   idxs  = (int*)(gates + R_ROWS);

    const int b    = blockIdx.x;
    const int tid  = threadIdx.x;
    const int wave = tid >> 5;
    const int lane = tid & 31;

    // ---- Stage token vector + row indices into LDS ----
    if (tid < R_ROWS) idxs[tid] = indices[(long long)b * R_ROWS + tid];
    for (int c = tid; c < D_IN; c += BLOCK)
        xs[c] = x[(long long)b * D_IN + c];
    __syncthreads();

    // ---- Kick off async global->LDS copy of the 32 out_weight rows ----
    // 64KB total = 4096 x 16B chunks; each thread issues 16 b128 async loads.
    // These DMA in the background while Phase A does the dot products.
    {
        const unsigned lds_base =
            (unsigned)(unsigned long long)(void*)vrows;  // wave-relative LDS byte addr
#pragma unroll
        for (int i = 0; i < 16; ++i) {
            const int g   = tid + i * BLOCK;   // 0..4095
            const int row = g >> 7;            // 128 chunks per 2KB row
            const int c16 = g & 127;           // 16B chunk within row
            const float*   gsrc = out_w + (long long)idxs[row] * D_OUT + c16 * 4;
            const unsigned ldst = lds_base + (unsigned)(row * (VSTRIDE * 4) + c16 * 16);
            asm volatile("global_load_async_to_lds_b128 %0, %1, off"
                         :: "v"(ldst), "v"(gsrc)
                         : "memory");
        }
    }

    // ---- Phase A: 64 length-512 dot products (32 key + 32 gate), 8/wave ----
    for (int s = 0; s < 8; ++s) {
        const int d   = wave * 8 + s;       // uniform per wave
        const int row = d & (R_ROWS - 1);
        const float* __restrict__ W = (d < R_ROWS) ? in_w : gate_w;
        const float* __restrict__ wr = W + (long long)idxs[row] * D_IN + lane * 16;
        const float* __restrict__ xr = xs + lane * 16;
        float sum = 0.0f;
#pragma unroll
        for (int q = 0; q < 4; ++q) {
            float4 wv = ((const float4*)wr)[q];
            float4 xv = ((const float4*)xr)[q];
            sum += wv.x * xv.x + wv.y * xv.y + wv.z * xv.z + wv.w * xv.w;
        }
        // wave32 butterfly reduction
#pragma unroll
        for (int off = 16; off > 0; off >>= 1)
            sum += __shfl_xor(sum, off, 32);
        if (lane == 0) {
            if (d < R_ROWS) coefs[row] = sum;
            else            gates[row] = sum;
        }
    }
    __syncthreads();

    // ---- Phase B: coef *= gelu_tanh(gate) ----
    if (tid < R_ROWS) {
        float g  = gates[tid];
        float g3 = g * g * g;
        float t  = tanhf(0.7978845608028654f * (g + 0.044715f * g3));
        coefs[tid] = coefs[tid] * (0.5f * g * (1.0f + t));
    }

    // Each wave waits for its own async copies, then barrier publishes LDS.
    asm volatile("s_wait_asynccnt 0x0" ::: "memory");
    __syncthreads();

    // ---- Phase C: out[b,:] = sum_r coef[r] * vrows[r,:]  via WMMA ----
    // A (16x4 f32): coef[r0..r0+3] replicated across M rows.
    //   Layout: lanes 0-15 hold K=0(v0),K=1(v1); lanes 16-31 hold K=2,K=3.
    // B (4x16 f32): vrows[r0+k][col0 + n], n = lane%16, K split across halves.
    // All 16 rows of C/D are identical; M=0 row = acc[0] in lanes 0-15.
    const int half = lane >> 4;   // 0 or 1
    const int n    = lane & 15;
    for (int t = wave; t < (D_OUT / 16); t += 8) {   // 4 column tiles per wave
        const int col0 = t * 16;
        v8f acc = {0.f, 0.f, 0.f, 0.f, 0.f, 0.f, 0.f, 0.f};
#pragma unroll
        for (int r0 = 0; r0 < R_ROWS; r0 += 4) {
            const int ra = r0 + 2 * half;
            v2f A, Bv;
            A[0]  = coefs[ra];
            A[1]  = coefs[ra + 1];
            Bv[0] = vrows[ra * VSTRIDE + col0 + n];
            Bv[1] = vrows[(ra + 1) * VSTRIDE + col0 + n];
            acc = __builtin_amdgcn_wmma_f32_16x16x4_f32(
                false, A, false, Bv, (short)0, acc, false, false);
        }
        if (lane < 16)
            out[(long long)b * D_OUT + col0 + lane] = acc[0];
    }
}

extern "C" void kernel_launch(void* const* d_in, const int* in_sizes, int n_in,
                              void* d_out, int out_size, void* d_ws, size_t ws_size,
                              hipStream_t stream) {
    const float* x      = (const float*)d_in[0];
    const int*   idx    = (const int*)d_in[1];
    const float* in_w   = (const float*)d_in[2];
    const float* gate_w = (const float*)d_in[3];
    const float* out_w  = (const float*)d_in[4];
    float*       out    = (float*)d_out;

    const int n_tok = in_sizes[0] / D_IN;   // 4*2048 = 8192 tokens
    const size_t smem_bytes = SMEM_F * sizeof(float);   // ~69 KB dynamic LDS

    sparse_gated_mlp_kernel<<<dim3(n_tok), dim3(BLOCK), smem_bytes, stream>>>(
        x, idx, in_w, gate_w, out_w, out, n_tok);
}